// PCENTransform_61083024884033
// MI455X (gfx1250) — compile-verified
//
#include <hip/hip_runtime.h>

// PCEN transform for E(32, 256, 4096) fp32.
// Pass 1: exact parallel EMA scan (affine-map wave32 scan) + PCEN pointwise,
//         E streamed via GLOBAL_LOAD_ASYNC_TO_LDS_B128 double-buffering when
//         the toolchain exposes it; writes pcen into d_out, reduces global
//         min/max into d_ws keys.
// Pass 2: in-place affine renormalization of d_out.

#define T_LEN  4096
#define ROWS   (32 * 256)        // 8192 independent IIR rows
#define WPB    8                 // waves per block (256 threads, wave32)
#define PCEN_EPS 1e-6f

// ---- hardware transcendentals: v_exp_f32 (2^x) / v_log_f32 (log2 x) ----
#if __has_builtin(__builtin_amdgcn_exp2f)
__device__ __forceinline__ float fexp2(float x) { return __builtin_amdgcn_exp2f(x); }
#else
__device__ __forceinline__ float fexp2(float x) { return exp2f(x); }
#endif
#if __has_builtin(__builtin_amdgcn_logf)
__device__ __forceinline__ float flog2(float x) { return __builtin_amdgcn_logf(x); }
#else
__device__ __forceinline__ float flog2(float x) { return log2f(x); }
#endif

// ---- gfx1250 async global->LDS copy (ASYNCcnt-tracked) ----
#define HAVE_ASYNC_LDS __has_builtin(__builtin_amdgcn_global_load_async_to_lds_b128)
#if HAVE_ASYNC_LDS
typedef int v4i __attribute__((vector_size(16)));
typedef __attribute__((address_space(1))) v4i* gp128_t;  // global source
typedef __attribute__((address_space(3))) v4i* lp128_t;  // LDS destination
#endif
#if __has_builtin(__builtin_amdgcn_s_wait_asynccnt)
#define WAIT_ASYNC(n) __builtin_amdgcn_s_wait_asynccnt(n)
#else
#define WAIT_ASYNC(n) asm volatile("s_wait_asynccnt %0" ::"i"(n) : "memory")
#endif

// ---- monotone float <-> uint key so integer atomics give float min/max ----
__device__ __forceinline__ unsigned f2key(float f) {
  unsigned u = __float_as_uint(f);
  return (u & 0x80000000u) ? ~u : (u | 0x80000000u);
}
__device__ __forceinline__ float key2f(unsigned k) {
  return (k & 0x80000000u) ? __uint_as_float(k & 0x7fffffffu)
                           : __uint_as_float(~k);
}

__global__ void pcen_init(unsigned* __restrict__ mm) {
  mm[0] = 0xFFFFFFFFu;  // running-min key
  mm[1] = 0u;           // running-max key
}

__global__ __launch_bounds__(256) void pcen_scan(
    const float* __restrict__ E, float* __restrict__ P,
    const float* __restrict__ sp, const float* __restrict__ ap,
    const float* __restrict__ dp, const float* __restrict__ rp,
    unsigned* __restrict__ mm) {
  const int lane = threadIdx.x & 31;
  const int wave = threadIdx.x >> 5;
  const int row  = blockIdx.x * WPB + wave;

  float s = sp[0];
  s = fminf(fmaxf(s, 0.0f), 1.0f);          // jnp.clip(s, 0, 1)
  const float k     = 1.0f - s;
  const float alpha = ap[0];                 // raw alpha/r, per reference note
  const float delta = dp[0];
  const float r     = rp[0];
  const float dr    = fexp2(r * flog2(delta));   // delta ** r
  const float k2 = k * k, k3 = k2 * k, k4 = k2 * k2;

  const float* e_row = E + (size_t)row * T_LEN;
  float*       p_row = P + (size_t)row * T_LEN;

  // Seeding carry with E[row,0] makes m0(lane0,iter0) = (k+s)*E0 = E0,
  // reproducing the reference's M[0] = E[0] boundary condition.
  float carry = e_row[0];
  float lmin  =  __builtin_inff();
  float lmax  = -__builtin_inff();

#if HAVE_ASYNC_LDS
  // Per-wave double buffer: 2 x 128 floats (512 B tiles), same-wave
  // producer/consumer -> no workgroup barrier needed.
  __shared__ __align__(16) float stage[2][WPB][128];
  auto issue_tile = [&](int t0, int buf) {
    __builtin_amdgcn_global_load_async_to_lds_b128(
        (gp128_t)(e_row + t0 + lane * 4),
        (lp128_t)(&stage[buf][wave][lane * 4]), /*imm_offset=*/0, /*cpol=*/0);
  };
  issue_tile(0, 0);
  issue_tile(128, 1);
#endif

  for (int t0 = 0; t0 < T_LEN; t0 += 128) {
    float4 e;
#if HAVE_ASYNC_LDS
    const int buf = (t0 >> 7) & 1;
    if (t0 + 128 < T_LEN) { WAIT_ASYNC(1); } else { WAIT_ASYNC(0); }
    asm volatile("" ::: "memory");           // keep LDS read after the wait
    e = *(const float4*)(&stage[buf][wave][lane * 4]);
    if (t0 + 256 < T_LEN) issue_tile(t0 + 256, buf);
#else
    e = *(const float4*)(e_row + t0 + lane * 4);
    if (t0 + 128 < T_LEN)                    // gfx1250 global_prefetch_b8
      __builtin_prefetch(e_row + t0 + 128 + lane * 4, 0, 0);
#endif

    // Lane-local affine map over its 4 consecutive time steps: m_out = A*m + B
    float Ai = k4;
    float Bi = s * fmaf(k3, e.x, fmaf(k2, e.y, fmaf(k, e.z, e.w)));

    // Inclusive wave32 Kogge-Stone scan composing affine maps (exact).
#pragma unroll
    for (int off = 1; off < 32; off <<= 1) {
      float Au = __shfl_up(Ai, off);
      float Bu = __shfl_up(Bi, off);
      if (lane >= off) { Bi = fmaf(Ai, Bu, Bi); Ai = Ai * Au; }
    }
    // Exclusive prefix -> incoming EMA state for this lane's 4 elements.
    const float Aex = __shfl_up(Ai, 1);
    const float Bex = __shfl_up(Bi, 1);
    const float m_in = (lane == 0) ? carry : fmaf(Aex, carry, Bex);

    const float m0 = fmaf(k, m_in, s * e.x);
    const float m1 = fmaf(k, m0,   s * e.y);
    const float m2 = fmaf(k, m1,   s * e.z);
    const float m3 = fmaf(k, m2,   s * e.w);

    auto pc = [&](float ev, float mv) -> float {
      // (E / (eps+M)^alpha + delta)^r - delta^r, via v_log_f32 / v_exp_f32
      const float ratio = ev * fexp2(-alpha * flog2(PCEN_EPS + mv));
      return fexp2(r * flog2(ratio + delta)) - dr;
    };
    const float p0 = pc(e.x, m0), p1 = pc(e.y, m1);
    const float p2 = pc(e.z, m2), p3 = pc(e.w, m3);

    float4 o; o.x = p0; o.y = p1; o.z = p2; o.w = p3;
    *(float4*)(p_row + t0 + lane * 4) = o;

    lmin = fminf(lmin, fminf(fminf(p0, p1), fminf(p2, p3)));
    lmax = fmaxf(lmax, fmaxf(fmaxf(p0, p1), fmaxf(p2, p3)));

    carry = __shfl(m3, 31);                  // broadcast end-of-tile EMA state
  }

  // wave32 min/max reduction, then block, then global atomics on keys.
#pragma unroll
  for (int off = 16; off >= 1; off >>= 1) {
    lmin = fminf(lmin, __shfl_down(lmin, off));
    lmax = fmaxf(lmax, __shfl_down(lmax, off));
  }
  __shared__ float smin[WPB], smax[WPB];
  if (lane == 0) { smin[wave] = lmin; smax[wave] = lmax; }
  __syncthreads();
  if (threadIdx.x == 0) {
    float bmin = smin[0], bmax = smax[0];
#pragma unroll
    for (int i = 1; i < WPB; ++i) {
      bmin = fminf(bmin, smin[i]);
      bmax = fmaxf(bmax, smax[i]);
    }
    atomicMin(&mm[0], f2key(bmin));
    atomicMax(&mm[1], f2key(bmax));
  }
}

__global__ __launch_bounds__(256) void pcen_norm(
    float* __restrict__ P, const unsigned* __restrict__ mm, int n4) {
  const int i = blockIdx.x * blockDim.x + threadIdx.x;
  if (i >= n4) return;
  const float pmin  = key2f(mm[0]);
  const float pmax  = key2f(mm[1]);
  const float scale = 2.0f / (pmax - pmin);
  float4 v = ((const float4*)P)[i];
  v.x = fmaf(v.x - pmin, scale, -1.0f);
  v.y = fmaf(v.y - pmin, scale, -1.0f);
  v.z = fmaf(v.z - pmin, scale, -1.0f);
  v.w = fmaf(v.w - pmin, scale, -1.0f);
  ((float4*)P)[i] = v;
}

extern "C" void kernel_launch(void* const* d_in, const int* in_sizes, int n_in,
                              void* d_out, int out_size, void* d_ws, size_t ws_size,
                              hipStream_t stream) {
  const float* E  = (const float*)d_in[0];
  const float* s  = (const float*)d_in[1];
  const float* a  = (const float*)d_in[2];
  const float* dl = (const float*)d_in[3];
  const float* r  = (const float*)d_in[4];
  float* out      = (float*)d_out;
  unsigned* mm    = (unsigned*)d_ws;   // 2 x u32 scratch, re-seeded each call

  (void)in_sizes; (void)n_in; (void)ws_size;

  pcen_init<<<1, 1, 0, stream>>>(mm);
  pcen_scan<<<ROWS / WPB, 256, 0, stream>>>(E, out, s, a, dl, r, mm);
  const int n4 = out_size / 4;
  pcen_norm<<<(n4 + 255) / 256, 256, 0, stream>>>(out, mm, n4);
}